// Correlation_5385888989617
// MI455X (gfx1250) — compile-verified
//
#include <hip/hip_runtime.h>

typedef __attribute__((ext_vector_type(2))) float v2f;
typedef __attribute__((ext_vector_type(4))) float v4f;
typedef __attribute__((ext_vector_type(8))) float v8f;

#define H_IN   384
#define W_IN   384
#define H_OUT  377
#define W_OUT  377
#define KH     8
#define KW     8
#define TILE_X 32                   // two 16-wide WMMA column groups
#define STRIP_Y 64
#define ROWS   (STRIP_Y + KH - 1)   // 71 input rows per strip
#define XCOLS  (TILE_X + KW)        // 40 input cols per tile

// Two-stage dynamic-filter cross-correlation:
//  Stage 1: R[y, r, x] = sum_c X[y, x+c] * K[r, c]   (WMMA f32 16x16x4, M=x, N=r, K=c)
//  Stage 2: out[y, x]  = sum_r R[y+r, r, x]
__global__ __launch_bounds__(256)
void corr_wmma_kernel(const float* __restrict__ x,
                      const float* __restrict__ k,
                      float* __restrict__ out) {
  __shared__ float XS[ROWS * XCOLS];            // input strip (zero-padded)  ~11.4 KB
  __shared__ float RS[ROWS * KH * TILE_X];      // row-correlation results    ~72.7 KB

  const int tid  = threadIdx.x;
  const int lane = tid & 31;                    // wave32
  const int wave = tid >> 5;                    // 0..7
  const int b    = blockIdx.z;
  const int x0   = blockIdx.x * TILE_X;
  const int y0   = blockIdx.y * STRIP_Y;

  // ---- stage 0: stage input strip into LDS, zero-padded at edges ----
  const float* xb = x + (size_t)b * H_IN * W_IN;
  for (int idx = tid; idx < ROWS * XCOLS; idx += 256) {
    int yy = idx / XCOLS;
    int xx = idx - yy * XCOLS;
    int gy = y0 + yy;
    int gx = x0 + xx;
    float v = 0.0f;
    if (gy < H_IN && gx < W_IN) v = xb[gy * W_IN + gx];
    XS[idx] = v;
  }

  // ---- build B operand (per-sample kernel, transposed; cols 8..15 zero) ----
  // B[kk, n] layout (4x16 f32): lane -> N = lane%16, VGPR j & lane-half -> K = j + 2*(lane>=16)
  const int m    = lane & 15;                   // M row of A / N col of B
  const int half = lane >> 4;                   // 0 or 1
  const int koff = half * 2;
  v2f bm1 = {0.f, 0.f}, bm2 = {0.f, 0.f};
  if (m < KH) {
    const float* kb = k + (size_t)b * (KH * KW) + m * KW;
    bm1.x = kb[0 + koff]; bm1.y = kb[1 + koff]; // taps c = 0..3
    bm2.x = kb[4 + koff]; bm2.y = kb[5 + koff]; // taps c = 4..7
  }

  __syncthreads();

  // ---- stage 1: row correlations via V_WMMA_F32_16X16X4_F32 ----
  // A[m, kk] layout (16x4 f32): lane -> M = lane%16, VGPR j & half -> K = j + 2*half
  for (int yy = wave; yy < ROWS; yy += 8) {     // wave-uniform loop, EXEC stays full
    const float* xr = &XS[yy * XCOLS];

    // column group 0: outputs x = 0..15
    v2f a1, a2;
    a1.x = xr[m + koff];
    a1.y = xr[m + koff + 1];
    a2.x = xr[m + koff + 4];
    a2.y = xr[m + koff + 5];
    v8f acc0 = {};
    acc0 = __builtin_amdgcn_wmma_f32_16x16x4_f32(false, a1, false, bm1, (short)0, acc0, false, false);
    acc0 = __builtin_amdgcn_wmma_f32_16x16x4_f32(false, a2, false, bm2, (short)0, acc0, false, false);

    // column group 1: outputs x = 16..31
    v2f a3, a4;
    a3.x = xr[16 + m + koff];
    a3.y = xr[16 + m + koff + 1];
    a4.x = xr[16 + m + koff + 4];
    a4.y = xr[16 + m + koff + 5];
    v8f acc1 = {};
    acc1 = __builtin_amdgcn_wmma_f32_16x16x4_f32(false, a3, false, bm1, (short)0, acc1, false, false);
    acc1 = __builtin_amdgcn_wmma_f32_16x16x4_f32(false, a4, false, bm2, (short)0, acc1, false, false);

    // D layout: VGPR j holds D[M = j + 8*half][N = lane%16]; keep only N = r < 8
    if (m < KH) {
      float* dst = &RS[(yy * KH + m) * TILE_X + half * 8];
      v4f lo0 = {acc0[0], acc0[1], acc0[2], acc0[3]};
      v4f hi0 = {acc0[4], acc0[5], acc0[6], acc0[7]};
      *(v4f*)(dst)          = lo0;              // x = 8*half + 0..3
      *(v4f*)(dst + 4)      = hi0;              // x = 8*half + 4..7
      v4f lo1 = {acc1[0], acc1[1], acc1[2], acc1[3]};
      v4f hi1 = {acc1[4], acc1[5], acc1[6], acc1[7]};
      *(v4f*)(dst + 16)     = lo1;              // x = 16 + 8*half + 0..3
      *(v4f*)(dst + 16 + 4) = hi1;              // x = 16 + 8*half + 4..7
    }
  }

  __syncthreads();

  // ---- stage 2: accumulate over kernel rows, masked global store ----
  float* ob = out + (size_t)b * H_OUT * W_OUT;
  const int xo = tid & 31;
  const int yb = tid >> 5;                      // 0..7
  for (int i = 0; i < STRIP_Y / 8; ++i) {
    int yo = yb + i * 8;
    float s = 0.f;
#pragma unroll
    for (int r = 0; r < KH; ++r)
      s += RS[((yo + r) * KH + r) * TILE_X + xo];
    int gy = y0 + yo, gx = x0 + xo;
    if (gy < H_OUT && gx < W_OUT) ob[gy * W_OUT + gx] = s;
  }
}

extern "C" void kernel_launch(void* const* d_in, const int* in_sizes, int n_in,
                              void* d_out, int out_size, void* d_ws, size_t ws_size,
                              hipStream_t stream) {
  (void)in_sizes; (void)n_in; (void)d_ws; (void)ws_size; (void)out_size;
  const float* x = (const float*)d_in[0];   // [128,384,384,1] fp32
  const float* k = (const float*)d_in[1];   // [128,8,8,1] fp32
  float* out = (float*)d_out;               // [128,377,377,1] fp32
  dim3 grid((W_OUT + TILE_X - 1) / TILE_X,    // 12
            (H_OUT + STRIP_Y - 1) / STRIP_Y,  // 6
            128);                             // batch
  corr_wmma_kernel<<<grid, dim3(256, 1, 1), 0, stream>>>(x, k, out);
}